// EdgeGeometricExtractor_19885698580771
// MI455X (gfx1250) — compile-verified
//
#include <hip/hip_runtime.h>
#include <math.h>

typedef float v2f __attribute__((ext_vector_type(2)));
typedef float v8f __attribute__((ext_vector_type(8)));

#define FS 12   // feats LDS row stride in floats (bank-conflict-free for b128)
#define HS 36   // hidden LDS row stride in floats

// One butterfly step within each 16-lane DPP row: v += lane_xor(v, MASK).
// ROW_XMASK dpp16 control = 0x160 | mask (gfx10+ / gfx1250). Pure VALU, no DS.
template <int CTRL>
__device__ __forceinline__ float dpp_add_x(float v) {
    int s = __builtin_bit_cast(int, v);
    int p = __builtin_amdgcn_update_dpp(0, s, CTRL, 0xf, 0xf, true);
    return v + __builtin_bit_cast(float, p);
}

// Sum across each 16-lane half of the wave32 (one WMMA C-layout matrix row).
__device__ __forceinline__ float redux16(float v) {
    v = dpp_add_x<0x161>(v);  // row_xmask:1
    v = dpp_add_x<0x162>(v);  // row_xmask:2
    v = dpp_add_x<0x164>(v);  // row_xmask:4
    v = dpp_add_x<0x168>(v);  // row_xmask:8
    return v;
}

// Branchless erf, Abramowitz-Stegun 7.1.26 (max abs err ~1.5e-7).
__device__ __forceinline__ float erf_fast(float x) {
    float ax = fabsf(x);
    float t  = __builtin_amdgcn_rcpf(1.0f + 0.3275911f * ax);
    float y  = t * (0.254829592f + t * (-0.284496736f + t * (1.421413741f +
               t * (-1.453152027f + t * 1.061405429f))));
    float r  = 1.0f - y * __expf(-ax * ax);
    return copysignf(r, x);
}

__device__ __forceinline__ float gelu_exact(float x) {
    return 0.5f * x * (1.0f + erf_fast(x * 0.70710678118654752f));
}

__global__ __launch_bounds__(256) void edge_geo_mlp_kernel(
    const float4* __restrict__ boxes,
    const long long* __restrict__ eidx,
    const float* __restrict__ ln0_g, const float* __restrict__ ln0_b,
    const float* __restrict__ w1,    const float* __restrict__ b1,
    const float* __restrict__ ln1_g, const float* __restrict__ ln1_b,
    const float* __restrict__ w2,    const float* __restrict__ b2,
    const float* __restrict__ ln2_g, const float* __restrict__ ln2_b,
    float* __restrict__ out,
    long long nE, long long nTiles)
{
    __shared__ float s_feat[8][32 * FS];  // per-wave 32 edges x 8 feats (padded)
    __shared__ float s_hid [8][16 * HS];  // per-wave 16 rows x 32 hidden (padded)

    const int lane = threadIdx.x & 31;
    const int wave = threadIdx.x >> 5;
    const int col  = lane & 15;          // N index within a 16-wide tile / A row
    const int kh   = (lane >> 4) * 2;    // K offset for A/B fragments (0 or 2)
    const int mh   = (lane >> 4) * 8;    // M offset for C rows (0 or 8)

    // ---- LN0 params (8 each) ----
    float g0[8], be0[8];
    {
        float4 a = ((const float4*)ln0_g)[0], b = ((const float4*)ln0_g)[1];
        g0[0]=a.x; g0[1]=a.y; g0[2]=a.z; g0[3]=a.w;
        g0[4]=b.x; g0[5]=b.y; g0[6]=b.z; g0[7]=b.w;
        float4 c = ((const float4*)ln0_b)[0], d = ((const float4*)ln0_b)[1];
        be0[0]=c.x; be0[1]=c.y; be0[2]=c.z; be0[3]=c.w;
        be0[4]=d.x; be0[5]=d.y; be0[6]=d.z; be0[7]=d.w;
    }

    // ---- W1 (8x32) fragments: B dual layout, lane=N, VGPR/half=K ----
    v2f w1f[2][2];
    #pragma unroll
    for (int kb = 0; kb < 2; ++kb)
        #pragma unroll
        for (int nb = 0; nb < 2; ++nb) {
            int krow = 4 * kb + kh;
            w1f[kb][nb][0] = w1[(krow    ) * 32 + 16 * nb + col];
            w1f[kb][nb][1] = w1[(krow + 1) * 32 + 16 * nb + col];
        }

    // ---- W2 (32x64) fragments ----
    v2f w2f[8][4];
    #pragma unroll
    for (int kb = 0; kb < 8; ++kb)
        #pragma unroll
        for (int nb = 0; nb < 4; ++nb) {
            int krow = 4 * kb + kh;
            w2f[kb][nb][0] = w2[(krow    ) * 64 + 16 * nb + col];
            w2f[kb][nb][1] = w2[(krow + 1) * 64 + 16 * nb + col];
        }

    // ---- per-lane bias / LN params along N ----
    float g1v[2], be1v[2], b1v[2];
    #pragma unroll
    for (int nb = 0; nb < 2; ++nb) {
        int n = 16 * nb + col;
        g1v[nb] = ln1_g[n]; be1v[nb] = ln1_b[n]; b1v[nb] = b1[n];
    }
    float g2v[4], be2v[4], b2v[4];
    #pragma unroll
    for (int nb = 0; nb < 4; ++nb) {
        int n = 16 * nb + col;
        g2v[nb] = ln2_g[n]; be2v[nb] = ln2_b[n]; b2v[nb] = b2[n];
    }

    for (long long tile = blockIdx.x; tile < nTiles; tile += gridDim.x) {
        const long long base = tile * 256 + (long long)wave * 32;
        long long e  = base + lane;
        long long ec = e < nE ? e : nE - 1;

        // ---- gather + geometric features (one edge per lane) ----
        long long si = eidx[ec];
        long long ti = eidx[nE + ec];
        float4 bs = boxes[si];
        float4 bt = boxes[ti];
        float scx = (bs.x + bs.z) * 0.5f, scy = (bs.y + bs.w) * 0.5f;
        float sw  = fmaxf(bs.z - bs.x, 1e-6f), sh = fmaxf(bs.w - bs.y, 1e-6f);
        float tcx = (bt.x + bt.z) * 0.5f, tcy = (bt.y + bt.w) * 0.5f;
        float tw  = fmaxf(bt.z - bt.x, 1e-6f), th = fmaxf(bt.w - bt.y, 1e-6f);
        float dx = tcx - scx, dy = tcy - scy;
        float f[8];
        f[0] = dx;
        f[1] = dy;
        f[2] = sqrtf(dx * dx + dy * dy + 1e-6f);
        f[3] = atan2f(dy, dx) * 0.3183098861837907f;
        f[4] = logf((tw * th) / (sw * sh) + 1e-6f);
        f[5] = logf((tw / th) / (sw / sh) + 1e-6f);
        f[6] = logf(tw / sw + 1e-6f);
        f[7] = logf(th / sh + 1e-6f);

        // LN over the 8 features (in registers)
        float mu = 0.f;
        #pragma unroll
        for (int i = 0; i < 8; ++i) mu += f[i];
        mu *= 0.125f;
        float va = 0.f;
        #pragma unroll
        for (int i = 0; i < 8; ++i) { float d = f[i] - mu; va += d * d; }
        va *= 0.125f;
        float rs = rsqrtf(va + 1e-5f);
        #pragma unroll
        for (int i = 0; i < 8; ++i) f[i] = (f[i] - mu) * rs * g0[i] + be0[i];

        // stage to LDS in row-major [edge][feat]
        float4* fp = (float4*)&s_feat[wave][lane * FS];
        fp[0] = make_float4(f[0], f[1], f[2], f[3]);
        fp[1] = make_float4(f[4], f[5], f[6], f[7]);

        #pragma unroll
        for (int mt = 0; mt < 2; ++mt) {
            // ---- GEMM1: (16x8) @ (8x32), f32 WMMA K=4 steps ----
            const float* fr = &s_feat[wave][(mt * 16 + col) * FS];
            v8f c0 = {}, c1 = {};
            #pragma unroll
            for (int kb = 0; kb < 2; ++kb) {
                v2f a;
                a[0] = fr[kb * 4 + kh];
                a[1] = fr[kb * 4 + kh + 1];
                c0 = __builtin_amdgcn_wmma_f32_16x16x4_f32(false, a, false, w1f[kb][0], 0, c0, false, false);
                c1 = __builtin_amdgcn_wmma_f32_16x16x4_f32(false, a, false, w1f[kb][1], 0, c1, false, false);
            }

            // ---- bias + LN(32) + exact GELU, write hidden to LDS ----
            #pragma unroll
            for (int r = 0; r < 8; ++r) {
                float x0 = c0[r] + b1v[0];
                float x1 = c1[r] + b1v[1];
                float s  = redux16(x0 + x1);
                float sq = redux16(x0 * x0 + x1 * x1);
                float m  = s * (1.f / 32.f);
                float vv = sq * (1.f / 32.f) - m * m;
                float rstd = rsqrtf(vv + 1e-5f);
                float h0 = (x0 - m) * rstd * g1v[0] + be1v[0];
                float h1 = (x1 - m) * rstd * g1v[1] + be1v[1];
                h0 = gelu_exact(h0);
                h1 = gelu_exact(h1);
                int ml = r + mh;
                s_hid[wave][ml * HS + col]      = h0;
                s_hid[wave][ml * HS + col + 16] = h1;
            }

            // ---- GEMM2: (16x32) @ (32x64) ----
            const float* hr = &s_hid[wave][col * HS];
            v8f o0 = {}, o1 = {}, o2 = {}, o3 = {};
            #pragma unroll
            for (int kb = 0; kb < 8; ++kb) {
                v2f a;
                a[0] = hr[kb * 4 + kh];
                a[1] = hr[kb * 4 + kh + 1];
                o0 = __builtin_amdgcn_wmma_f32_16x16x4_f32(false, a, false, w2f[kb][0], 0, o0, false, false);
                o1 = __builtin_amdgcn_wmma_f32_16x16x4_f32(false, a, false, w2f[kb][1], 0, o1, false, false);
                o2 = __builtin_amdgcn_wmma_f32_16x16x4_f32(false, a, false, w2f[kb][2], 0, o2, false, false);
                o3 = __builtin_amdgcn_wmma_f32_16x16x4_f32(false, a, false, w2f[kb][3], 0, o3, false, false);
            }

            // ---- bias + LN(64) + store (lanes 0-15 cover a contiguous 64B row seg) ----
            #pragma unroll
            for (int r = 0; r < 8; ++r) {
                float x0 = o0[r] + b2v[0];
                float x1 = o1[r] + b2v[1];
                float x2 = o2[r] + b2v[2];
                float x3 = o3[r] + b2v[3];
                float s  = redux16(x0 + x1 + x2 + x3);
                float sq = redux16(x0 * x0 + x1 * x1 + x2 * x2 + x3 * x3);
                float m  = s * (1.f / 64.f);
                float vv = sq * (1.f / 64.f) - m * m;
                float rstd = rsqrtf(vv + 1e-5f);
                long long erow = base + mt * 16 + r + mh;
                if (erow < nE) {
                    float* op = out + erow * 64 + col;
                    op[0]  = (x0 - m) * rstd * g2v[0] + be2v[0];
                    op[16] = (x1 - m) * rstd * g2v[1] + be2v[1];
                    op[32] = (x2 - m) * rstd * g2v[2] + be2v[2];
                    op[48] = (x3 - m) * rstd * g2v[3] + be2v[3];
                }
            }
        }
    }
}

extern "C" void kernel_launch(void* const* d_in, const int* in_sizes, int n_in,
                              void* d_out, int out_size, void* d_ws, size_t ws_size,
                              hipStream_t stream) {
    const float4*    boxes = (const float4*)d_in[0];
    const long long* eidx  = (const long long*)d_in[1];
    const float* ln0_g = (const float*)d_in[2];
    const float* ln0_b = (const float*)d_in[3];
    const float* w1    = (const float*)d_in[4];
    const float* b1    = (const float*)d_in[5];
    const float* ln1_g = (const float*)d_in[6];
    const float* ln1_b = (const float*)d_in[7];
    const float* w2    = (const float*)d_in[8];
    const float* b2    = (const float*)d_in[9];
    const float* ln2_g = (const float*)d_in[10];
    const float* ln2_b = (const float*)d_in[11];
    float* out = (float*)d_out;

    long long nE = (long long)in_sizes[1] / 2;   // edge_index is (2, N_EDGES)
    long long nTiles = (nE + 255) / 256;         // 256 edges per block per iter
    long long grid = nTiles < 1250 ? nTiles : 1250;

    edge_geo_mlp_kernel<<<(int)grid, 256, 0, stream>>>(
        boxes, eidx, ln0_g, ln0_b, w1, b1, ln1_g, ln1_b, w2, b2, ln2_g, ln2_b,
        out, nE, nTiles);
}